// ErbFilterBank_5463198401034
// MI455X (gfx1250) — compile-verified
//
#include <hip/hip_runtime.h>

typedef float v2f __attribute__((ext_vector_type(2)));
typedef float v4f __attribute__((ext_vector_type(4)));
typedef float v8f __attribute__((ext_vector_type(8)));

#define T_LEN  160000
#define N_CH   64
#define N_B    4
#define CHUNK  16000
#define WARM   2048
#define NCHUNK (T_LEN / CHUNK)

// D(16x16,f32) = A(16x4,f32) x B(4x16,f32) + C  -- CDNA5 fp32 WMMA
static __device__ __forceinline__ v8f wmma_f32_16x16x4(v2f a, v2f b, v8f c) {
  return __builtin_amdgcn_wmma_f32_16x16x4_f32(false, a, false, b, (short)0, c,
                                               false, false);
}

// Wave-uniform broadcast from a fixed lane: v_readlane_b32 (no LDS traffic).
static __device__ __forceinline__ float rdlane(float v, int srcLane) {
  return __int_as_float(__builtin_amdgcn_readlane(__float_as_int(v), srcLane));
}

static __device__ __forceinline__ int iclamp(int v, int lo, int hi) {
  return v < lo ? lo : (v > hi ? hi : v);
}

__global__ __launch_bounds__(32)
void erb_gammatone_kernel(const float* __restrict__ x,
                          const float* __restrict__ sos,
                          float* __restrict__ out)
{
  __shared__ alignas(16) float bufA[256];
  __shared__ alignas(16) float bufB[256];

  const int lane = threadIdx.x & 31;
  const int half = lane >> 4;   // which 16-lane half
  const int lid  = lane & 15;

  const int wg    = blockIdx.x;
  const int chunk = wg % NCHUNK;
  const int pair  = wg / NCHUNK;
  const int c     = pair % N_CH;
  const int b     = pair / N_CH;

  // sos: (64,3,5) row-major. Shared denominator (1,a1,a2); per-stage (b0,b1).
  const float* sc = sos + c * 15;
  const float a1 = sc[1*5 + 4];
  const float a2 = sc[2*5 + 4];
  float b0c[4], b1c[4];
  #pragma unroll
  for (int k = 0; k < 4; ++k) { b0c[k] = sc[k]; b1c[k] = sc[5 + k]; }

  // Build per-lane WMMA operands from the impulse response h[n]:
  //   L  (16x16 lower-tri Toeplitz, h[M-K])  -> four 16x4 A-slices (v2f each)
  //   Hp (16x4: cols u[n]=h[n+1], v[n]=-a2*h[n], 0, 0) -> one v2f
  // A-matrix layout (f32 16x4): lanes0-15 M=lid K={4s+r}, lanes16-31 M=lid K={4s+r+2}
  float aL[4][2];
  #pragma unroll
  for (int s = 0; s < 4; ++s) { aL[s][0] = 0.f; aL[s][1] = 0.f; }
  float hp0 = 0.f, hp1 = 0.f;
  float h14 = 0.f, h15 = 0.f, h16 = 0.f;
  {
    float h_nm1 = 0.f;
    float h_n   = 1.f;                 // h[0]
    for (int n = 0; n <= 16; ++n) {
      #pragma unroll
      for (int s = 0; s < 4; ++s) {
        #pragma unroll
        for (int r = 0; r < 2; ++r) {
          const int K = 4*s + r + 2*half;
          if (lid - K == n) aL[s][r] = h_n;
        }
      }
      if (half == 0) {
        if (n == lid + 1) hp0 = h_n;        // u[lid]
        if (n == lid)     hp1 = -a2 * h_n;  // v[lid]
      }
      if (n == 14) h14 = h_n;
      if (n == 15) h15 = h_n;
      if (n == 16) h16 = h_n;
      const float h_np1 = -a1 * h_n - a2 * h_nm1;
      h_nm1 = h_n; h_n = h_np1;
    }
  }
  const float u15 = h16, v15 = -a2 * h15;   // block-boundary 2x2 transition G
  const float u14 = h15, v14 = -a2 * h14;

  v2f A0, A1, A2, A3, Hp;
  A0[0]=aL[0][0]; A0[1]=aL[0][1];
  A1[0]=aL[1][0]; A1[1]=aL[1][1];
  A2[0]=aL[2][0]; A2[1]=aL[2][1];
  A3[0]=aL[3][0]; A3[1]=aL[3][1];
  Hp[0]=hp0; Hp[1]=hp1;

  const float* xb = x + (size_t)b * T_LEN;
  float*       ob = out + ((size_t)b * N_CH + c) * T_LEN;

  const int t0 = chunk * CHUNK;
  const int tw = (chunk == 0) ? 0 : (t0 - WARM);   // zero-state warm-up pre-roll
  const int ngroups = ((t0 + CHUNK - tw) + 255) >> 8;

  float st1[4] = {0.f,0.f,0.f,0.f};   // per-stage carried y[-1]
  float st2[4] = {0.f,0.f,0.f,0.f};   // per-stage carried y[-2]
  float fp [4] = {0.f,0.f,0.f,0.f};   // per-stage carried previous input sample

  for (int g = 0; g < ngroups; ++g) {
    const int tg = tw + g * 256;      // group = 16 blocks x 16 samples

    // Stage-0 forcing F (B-matrix layout: Vr lanes0-15 row 4s+r, lanes16-31 row 4s+r+2).
    // Unguarded, address-clamped loads (no exec-mask branches): samples past T_LEN
    // only exist in the discarded tail overhang; x[-1] fixed up with one cndmask.
    v2f Bf[4];
    #pragma unroll
    for (int s = 0; s < 4; ++s) {
      const int base = tg + 16*lid + 4*s + 2*half;
      const float xm1r = xb[iclamp(base - 1, 0, T_LEN - 1)];
      const float x0   = xb[iclamp(base,     0, T_LEN - 1)];
      const float x1   = xb[iclamp(base + 1, 0, T_LEN - 1)];
      const float xm1  = (base >= 1) ? xm1r : 0.f;
      Bf[s][0] = b0c[0]*x0 + b1c[0]*xm1;
      Bf[s][1] = b0c[0]*x1 + b1c[0]*x0;
    }

    float pend = 0.f;                 // last output sample of previous stage (this group)
    #pragma unroll
    for (int k = 0; k < 4; ++k) {
      if (k > 0) {
        // Forcing from previous stage's output held in LDS: f[n]=b0*y[n]+b1*y[n-1]
        const float* src = (k == 2) ? bufB : bufA;
        const float fpk = fp[k];
        #pragma unroll
        for (int s = 0; s < 4; ++s) {
          #pragma unroll
          for (int r = 0; r < 2; ++r) {
            const int m  = 4*s + r + 2*half;
            const int lt = 16*lid + m;
            const float cur = src[lt];
            const float prv = (lt > 0) ? src[lt-1] : fpk;
            Bf[s][r] = b0c[k]*cur + b1c[k]*prv;
          }
        }
        fp[k] = pend;                 // for next group
      }

      // P = L * F : 16x16x16 fp32 matmul via 4 K-slices of v_wmma_f32_16x16x4_f32
      v8f acc = {};
      acc = wmma_f32_16x16x4(A0, Bf[0], acc);
      acc = wmma_f32_16x16x4(A1, Bf[1], acc);
      acc = wmma_f32_16x16x4(A2, Bf[2], acc);
      acc = wmma_f32_16x16x4(A3, Bf[3], acc);

      // Sequential 2x2 affine scan over the 16 blocks (rows 14/15 live in
      // acc[6]/acc[7] on lanes 16..31 per the C/D layout). readlane broadcasts.
      float s1 = st1[k], s2 = st2[k];
      float q0 = 0.f, q1 = 0.f;       // S matrix entries for this lane's column
      #pragma unroll
      for (int j = 0; j < 16; ++j) {
        const float p15 = rdlane(acc[7], 16 + j);
        const float p14 = rdlane(acc[6], 16 + j);
        if (lane == j) { q0 = s1; q1 = s2; }
        const float y15n = fmaf(u15, s1, fmaf(v15, s2, p15));
        const float y14n = fmaf(u14, s1, fmaf(v14, s2, p14));
        s1 = y15n; s2 = y14n;
      }
      st1[k] = s1; st2[k] = s2;

      // Y = Hp(16x4) * S(4x16) + P : one more WMMA applies boundary correction
      v2f SB; SB[0] = q0; SB[1] = q1;  // lanes>=16 stay 0 (padded K rows)
      acc = wmma_f32_16x16x4(Hp, SB, acc);

      pend = rdlane(acc[7], 31);       // Y[15] of last block

      // Stash Y in LDS (sample-major): stage routing + linearized global store
      __syncthreads();
      float* dst = (k == 0 || k == 2) ? bufA : bufB;
      #pragma unroll
      for (int r = 0; r < 8; ++r) {
        const int lt = 16*lid + r + 8*half;
        dst[lt] = acc[r];
      }
      __syncthreads();

      if (k == 3) {
        // Coalesced non-temporal b128 stores; skip warm-up and tail overhang.
        #pragma unroll
        for (int q = 0; q < 2; ++q) {
          const int ltv = lane*8 + 4*q;
          const int t   = tg + ltv;
          if (t >= t0 && t < t0 + CHUNK) {
            const v4f v = *(const v4f*)&bufB[ltv];
            __builtin_nontemporal_store(v, (v4f*)&ob[t]);
          }
        }
      }
    }
  }
}

extern "C" void kernel_launch(void* const* d_in, const int* in_sizes, int n_in,
                              void* d_out, int out_size, void* d_ws, size_t ws_size,
                              hipStream_t stream) {
  (void)in_sizes; (void)n_in; (void)out_size; (void)d_ws; (void)ws_size;
  const float* x   = (const float*)d_in[0];
  const float* sos = (const float*)d_in[1];
  float* out = (float*)d_out;

  dim3 grid(N_B * N_CH * NCHUNK);   // 2560 single-wave workgroups
  dim3 block(32);
  hipLaunchKernelGGL(erb_gammatone_kernel, grid, block, 0, stream, x, sos, out);
}